// InfoNCELoss_43422119363030
// MI455X (gfx1250) — compile-verified
//
#include <hip/hip_runtime.h>

#define N_TOTAL 8192
#define HALF    4096
#define D       256
#define INV_T   (1.0f / 0.07f)
#define NSPLIT  8
#define TILES_PER_SPLIT (N_TOTAL / 16 / NSPLIT)   // 64 column tiles per split
#define LDS_STRIDE 260                             // 256 + 4 pad: 4*r bank spread

typedef __attribute__((ext_vector_type(2))) float v2f;
typedef __attribute__((ext_vector_type(8))) float v8f;

// ---------------- zero per-row accumulators ----------------
__global__ void k_zero(float* __restrict__ sumexp, int* __restrict__ rankc) {
    int i = blockIdx.x * blockDim.x + threadIdx.x;
    if (i < N_TOTAL) { sumexp[i] = 0.0f; rankc[i] = 0; }
}

// ---------------- row L2-normalize: xn = x / max(||x||, eps) ----------------
__global__ void k_normalize(const float* __restrict__ feats,
                            const float* __restrict__ feats_prime,
                            float* __restrict__ xn) {
    int row = blockIdx.x;
    int t   = threadIdx.x;                       // 256 threads = D
    const float* src = (row < HALF) ? (feats + (size_t)row * D)
                                    : (feats_prime + (size_t)(row - HALF) * D);
    float v = src[t];
    __shared__ float red[256];
    red[t] = v * v;
    __syncthreads();
    for (int s = 128; s > 0; s >>= 1) {
        if (t < s) red[t] += red[t + s];
        __syncthreads();
    }
    float norm = fmaxf(sqrtf(red[0]), 1e-8f);
    xn[(size_t)row * D + t] = v / norm;
}

// ---------------- positive-pair cosine per row (one wave per row) ----------------
__global__ void k_possim(const float* __restrict__ xn, float* __restrict__ pos) {
    int gtid = blockIdx.x * blockDim.x + threadIdx.x;
    int wave = gtid >> 5;                        // 8192 waves == rows
    int lane = gtid & 31;
    int row  = wave;
    int prow = (row + HALF) & (N_TOTAL - 1);
    float acc = 0.0f;
    for (int k = lane; k < D; k += 32)
        acc += xn[(size_t)row * D + k] * xn[(size_t)prow * D + k];
    for (int m = 16; m > 0; m >>= 1) acc += __shfl_xor(acc, m, 32);
    if (lane == 0) pos[row] = acc;               // raw cosine
}

// ---------------- main: streamed cos tiles via f32 WMMA + LDS-shared B ----------
// 1 wave owns 16 rows; the 8 waves of a block share each 16-col B tile via LDS.
__global__ void __launch_bounds__(256)
k_simtile(const float* __restrict__ xn, const float* __restrict__ pos,
          float* __restrict__ sumexp, int* __restrict__ rankc) {
    __shared__ float Bbuf[2][16 * LDS_STRIDE];   // 2 x 16.6 KB double buffer

    const int tid       = threadIdx.x;
    const int waveInBlk = tid >> 5;              // 8 waves / block
    const int lane      = tid & 31;
    const int laneHalf  = lane >> 4;             // 0: lanes 0-15, 1: lanes 16-31
    const int lmod      = lane & 15;
    const int rowBase   = (blockIdx.x * 8 + waveInBlk) * 16;
    const int colTile0  = blockIdx.y * TILES_PER_SPLIT;

    // staging mapping: thread copies 16 floats of row (tid%16), segment (tid/16)
    const int sr   = tid & 15;
    const int sseg = tid >> 4;

    // A fragment: lane holds row rowBase+lmod, K pair 2*laneHalf+{0,1} per step
    const int rowA = rowBase + lmod;
    const int kOff = laneHalf * 2;

    v2f afrag[64];
#pragma unroll
    for (int kk = 0; kk < 64; ++kk)
        afrag[kk] = *(const v2f*)(xn + (size_t)rowA * D + kk * 4 + kOff);

    // per-lane: the 8 rows this lane's C registers cover
    float posv[8];
    int   posj[8];
#pragma unroll
    for (int v = 0; v < 8; ++v) {
        int i   = rowBase + v + 8 * laneHalf;
        posv[v] = pos[i];
        posj[v] = (i + HALF) & (N_TOTAL - 1);
    }

    float se[8];
    int   ct[8];
#pragma unroll
    for (int v = 0; v < 8; ++v) { se[v] = 0.0f; ct[v] = 0; }

    float4 pre[4];

    // prologue: stage tile 0 into buffer 0
    {
        int colBase = colTile0 * 16;
        const float4* src = (const float4*)(xn + (size_t)(colBase + sr) * D + sseg * 16);
#pragma unroll
        for (int q = 0; q < 4; ++q) pre[q] = src[q];
        float4* dst = (float4*)&Bbuf[0][sr * LDS_STRIDE + sseg * 16];
#pragma unroll
        for (int q = 0; q < 4; ++q) dst[q] = pre[q];
    }
    __syncthreads();

    for (int t = 0; t < TILES_PER_SPLIT; ++t) {
        const int buf      = t & 1;
        const bool haveNxt = (t + 1) < TILES_PER_SPLIT;

        // prefetch next B tile into registers (latency hides under WMMA chain)
        if (haveNxt) {
            int colBase = (colTile0 + t + 1) * 16;
            const float4* src = (const float4*)(xn + (size_t)(colBase + sr) * D + sseg * 16);
#pragma unroll
            for (int q = 0; q < 4; ++q) pre[q] = src[q];
        }

        // compute current tile: B fragments from LDS (bank-conflict-free b64)
        const float* bbase = &Bbuf[buf][lmod * LDS_STRIDE + kOff];
        v8f c = {};
#pragma unroll
        for (int kk = 0; kk < 64; ++kk) {
            v2f b = *(const v2f*)(bbase + kk * 4);
            c = __builtin_amdgcn_wmma_f32_16x16x4_f32(
                    false, afrag[kk], false, b, (short)0, c, false, false);
        }

        // C layout: VGPR v -> row (v + 8*laneHalf), column = lane%16
        const int colBase = (colTile0 + t) * 16;
        const int j = colBase + lmod;
#pragma unroll
        for (int v = 0; v < 8; ++v) {
            int   i  = rowBase + v + 8 * laneHalf;
            float cv = c[v];
            if (j != i) {                         // diagonal masked to -inf
                se[v] += expf(cv * INV_T);        // logsumexp includes positive col
                if (j != posj[v] && cv > posv[v]) // rank: strictly-greater, pos masked
                    ct[v]++;
            }
        }

        // commit prefetched tile into the other buffer
        if (haveNxt) {
            float4* dst = (float4*)&Bbuf[buf ^ 1][sr * LDS_STRIDE + sseg * 16];
#pragma unroll
            for (int q = 0; q < 4; ++q) dst[q] = pre[q];
        }
        __syncthreads();
    }

    // reduce each row across its 16 lanes (shfl within 16-lane groups)
#pragma unroll
    for (int v = 0; v < 8; ++v) {
        float s  = se[v];
        int   c2 = ct[v];
        for (int m = 8; m > 0; m >>= 1) {
            s  += __shfl_xor(s,  m, 16);
            c2 += __shfl_xor(c2, m, 16);
        }
        if (lmod == 0) {
            int i = rowBase + v + 8 * laneHalf;
            atomicAdd(&sumexp[i], s);
            atomicAdd(&rankc[i],  c2);
        }
    }
}

// ---------------- final reduction to 4 scalars ----------------
__global__ void k_reduce(const float* __restrict__ pos,
                         const float* __restrict__ sumexp,
                         const int*   __restrict__ rankc,
                         float* __restrict__ out) {
    __shared__ float s_nll[256], s_a1[256], s_a5[256], s_rs[256];
    int t = threadIdx.x;
    float nll = 0.0f, a1 = 0.0f, a5 = 0.0f, rs = 0.0f;
    for (int i = t; i < N_TOTAL; i += 256) {
        nll += -pos[i] * INV_T + logf(sumexp[i]);
        int r = rankc[i];
        a1 += (r == 0) ? 1.0f : 0.0f;
        a5 += (r < 5)  ? 1.0f : 0.0f;
        rs += (float)r;
    }
    s_nll[t] = nll; s_a1[t] = a1; s_a5[t] = a5; s_rs[t] = rs;
    __syncthreads();
    for (int s = 128; s > 0; s >>= 1) {
        if (t < s) {
            s_nll[t] += s_nll[t + s];
            s_a1[t]  += s_a1[t + s];
            s_a5[t]  += s_a5[t + s];
            s_rs[t]  += s_rs[t + s];
        }
        __syncthreads();
    }
    if (t == 0) {
        float inv = 1.0f / (float)N_TOTAL;
        out[0] = s_nll[0] * inv;
        out[1] = s_a1[0] * inv;
        out[2] = s_a5[0] * inv;
        out[3] = 1.0f + s_rs[0] * inv;
    }
}

extern "C" void kernel_launch(void* const* d_in, const int* in_sizes, int n_in,
                              void* d_out, int out_size, void* d_ws, size_t ws_size,
                              hipStream_t stream) {
    const float* feats       = (const float*)d_in[0];
    const float* feats_prime = (const float*)d_in[1];
    float* out = (float*)d_out;

    // workspace layout: xn[8192*256] | pos[8192] | sumexp[8192] | rankc[8192]
    float* xn     = (float*)d_ws;
    float* pos    = xn + (size_t)N_TOTAL * D;
    float* sumexp = pos + N_TOTAL;
    int*   rankc  = (int*)(sumexp + N_TOTAL);

    k_zero<<<(N_TOTAL + 255) / 256, 256, 0, stream>>>(sumexp, rankc);
    k_normalize<<<N_TOTAL, 256, 0, stream>>>(feats, feats_prime, xn);
    k_possim<<<(N_TOTAL * 32) / 256, 256, 0, stream>>>(xn, pos);
    k_simtile<<<dim3(N_TOTAL / (8 * 16), NSPLIT), 256, 0, stream>>>(xn, pos, sumexp, rankc);
    k_reduce<<<1, 256, 0, stream>>>(pos, sumexp, rankc, out);
}